// GTLayer_9500467659500
// MI455X (gfx1250) — compile-verified
//
#include <hip/hip_runtime.h>
#include <math.h>

#define NND   100000
#define EMBED 64
#define QKVC  192
#define FFND  256
#define LNEPS 1e-5f

typedef float v2f __attribute__((ext_vector_type(2)));
typedef float v8f __attribute__((ext_vector_type(8)));

__device__ __forceinline__ v8f wmma4(v2f a, v2f b, v8f c) {
  // V_WMMA_F32_16X16X4_F32 : D = A(16x4) * B(4x16) + C(16x16), fp32
  return __builtin_amdgcn_wmma_f32_16x16x4_f32(false, a, false, b, (short)0, c,
                                               false, false);
}

// CDNA5 async bulk copy global -> LDS (tracked by ASYNCcnt)
__device__ __forceinline__ void async_copy_b128(unsigned lds_byte_addr,
                                                const void* gaddr) {
  asm volatile("global_load_async_to_lds_b128 %0, %1, off"
               :: "v"(lds_byte_addr), "v"(gaddr)
               : "memory");
}
__device__ __forceinline__ void wait_async0() {
  asm volatile("s_wait_asynccnt 0" ::: "memory");
}
// truncate generic LDS pointer to its 32-bit LDS byte address
__device__ __forceinline__ unsigned lds_addr_of(const void* p) {
  return (unsigned)(unsigned long long)p;
}

// ---------------------------------------------------------------- zero int scratch
__global__ __launch_bounds__(256) void k_zero_int(int* __restrict__ p, int total) {
  int i = blockIdx.x * 256 + threadIdx.x;
  if (i < total) p[i] = 0;
}

// ---------------------------------------------------------------- qkv = x @ W + b
// one 16x16 output tile per wave; K=64 -> 16 wmma per tile
__global__ __launch_bounds__(256) void k_qkv(const float* __restrict__ x,
                                             const float* __restrict__ w,
                                             const float* __restrict__ bias,
                                             float* __restrict__ qkv) {
  const int lane = threadIdx.x & 31;
  const int wv   = threadIdx.x >> 5;
  const int tile = blockIdx.x * 8 + wv;     // 9375 blocks * 8 waves = 75000 tiles
  const int tm   = tile / 12;
  const int tn   = tile % 12;
  const int r    = lane & 15;
  const int half = lane >> 4;               // 0: K pair {k,k+1}, 1: {k+2,k+3}
  const int koff = half * 2;

  const float* arow = x + (size_t)(tm * 16 + r) * EMBED + koff;
  const float* bcol = w + (size_t)koff * QKVC + tn * 16 + r;

  v8f acc = {};
#pragma unroll
  for (int k0 = 0; k0 < EMBED; k0 += 4) {
    v2f a; a.x = arow[k0];                   a.y = arow[k0 + 1];
    v2f b; b.x = bcol[(size_t)k0 * QKVC];    b.y = bcol[(size_t)(k0 + 1) * QKVC];
    acc = wmma4(a, b, acc);
  }
  const float bb = bias[tn * 16 + r];
  float* outp = qkv + (size_t)(tm * 16 + half * 8) * QKVC + tn * 16 + r;
#pragma unroll
  for (int i = 0; i < 8; ++i) outp[(size_t)i * QKVC] = acc[i] + bb;
}

// ---------------------------------------------------------------- CSR build
__global__ __launch_bounds__(256) void k_count(const int* __restrict__ eidx,
                                               int nedges, int* __restrict__ counts) {
  int e = blockIdx.x * 256 + threadIdx.x;
  if (e < nedges) atomicAdd(&counts[eidx[e]], 1);
}

// single-block exclusive scan of counts[n] -> base[n]
__global__ __launch_bounds__(1024) void k_scan(const int* __restrict__ counts,
                                               int* __restrict__ base, int n) {
  __shared__ int psum[1024];
  const int t = threadIdx.x;
  const int chunk = (n + 1023) / 1024;
  const int s0 = t * chunk;
  const int s1 = (s0 + chunk < n) ? (s0 + chunk) : n;
  int s = 0;
  for (int i = s0; i < s1; ++i) s += counts[i];
  psum[t] = s;
  __syncthreads();
  for (int off = 1; off < 1024; off <<= 1) {
    int v = (t >= off) ? psum[t - off] : 0;
    __syncthreads();
    psum[t] += v;
    __syncthreads();
  }
  int pre = (t == 0) ? 0 : psum[t - 1];
  for (int i = s0; i < s1; ++i) { base[i] = pre; pre += counts[i]; }
}

__global__ __launch_bounds__(256) void k_scatter(const int* __restrict__ eidx,
                                                 int nedges,
                                                 const int* __restrict__ base,
                                                 int* __restrict__ cursor,
                                                 int* __restrict__ elist) {
  int e = blockIdx.x * 256 + threadIdx.x;
  if (e < nedges) {
    int row = eidx[e];
    int pos = atomicAdd(&cursor[row], 1);
    elist[base[row] + pos] = e;
  }
}

// ---------------------------------------------------------------- gather attention
// one thread per (node, head): no float atomics at all
__global__ __launch_bounds__(256) void k_attn(const int* __restrict__ eidx,
                                              int nedges,
                                              const int* __restrict__ counts,
                                              const int* __restrict__ base,
                                              const int* __restrict__ elist,
                                              const float* __restrict__ qkv,
                                              float* __restrict__ acc) {
  int gid = blockIdx.x * 256 + threadIdx.x;      // N*8 threads
  if (gid >= NND * 8) return;
  int node = gid >> 3, h = gid & 7;

  float q[8];
  const float* qp = qkv + (size_t)node * QKVC + h * 24;
#pragma unroll
  for (int d = 0; d < 8; ++d) q[d] = qp[d];

  const int start = base[node];
  const int deg   = counts[node];

  float mx = -__builtin_huge_valf();
  for (int j = 0; j < deg; ++j) {
    int e = elist[start + j];
    const float* kp = qkv + (size_t)eidx[nedges + e] * QKVC + h * 24 + 8;
    float dot = 0.0f;
#pragma unroll
    for (int d = 0; d < 8; ++d) dot += q[d] * kp[d];
    mx = fmaxf(mx, dot * 0.125f);                 // scale = EMBED^-0.5
  }

  float denom = 0.0f;
  float o[8] = {0.f, 0.f, 0.f, 0.f, 0.f, 0.f, 0.f, 0.f};
  for (int j = 0; j < deg; ++j) {
    int e = elist[start + j];
    const float* kp = qkv + (size_t)eidx[nedges + e] * QKVC + h * 24 + 8;
    float dot = 0.0f;
#pragma unroll
    for (int d = 0; d < 8; ++d) dot += q[d] * kp[d];
    float ex = __expf(dot * 0.125f - mx);
    denom += ex;
    const float* vp = kp + 8;
#pragma unroll
    for (int d = 0; d < 8; ++d) o[d] += ex * vp[d];
  }

  float inv = (deg > 0) ? 1.0f / denom : 0.0f;
  float* op = acc + (size_t)node * EMBED + h * 8;
#pragma unroll
  for (int d = 0; d < 8; ++d) op[d] = o[d] * inv;
}

// ---------------------------------------------------------------- LN1: h1 = LN(x + attn)
__global__ __launch_bounds__(256) void k_ln1(const float* __restrict__ x,
                                             const float* __restrict__ acc,
                                             const float* __restrict__ g,
                                             const float* __restrict__ b,
                                             float* __restrict__ h1) {
  int lane = threadIdx.x & 31;
  int row  = blockIdx.x * 8 + (threadIdx.x >> 5);
  const float* xr = x + (size_t)row * EMBED;
  const float* ar = acc + (size_t)row * EMBED;
  int c0 = lane * 2;
  float y0 = xr[c0]     + ar[c0];
  float y1 = xr[c0 + 1] + ar[c0 + 1];
  float s = y0 + y1;
#pragma unroll
  for (int off = 16; off > 0; off >>= 1) s += __shfl_xor(s, off);
  float mean = s * (1.0f / 64.0f);
  float d0 = y0 - mean, d1 = y1 - mean;
  float v = d0 * d0 + d1 * d1;
#pragma unroll
  for (int off = 16; off > 0; off >>= 1) v += __shfl_xor(v, off);
  float rs = rsqrtf(v * (1.0f / 64.0f) + LNEPS);
  float* o = h1 + (size_t)row * EMBED;
  o[c0]     = d0 * rs * g[c0]     + b[c0];
  o[c0 + 1] = d1 * rs * g[c0 + 1] + b[c0 + 1];
}

// ---------------------------------------------------------------- fused FFN + LN2
// weights staged into LDS via CDNA5 async copies; all GEMM math on WMMA
__global__ __launch_bounds__(256) void k_ffn(float* __restrict__ h1out,
                                             const float* __restrict__ w1,
                                             const float* __restrict__ b1,
                                             const float* __restrict__ w2,
                                             const float* __restrict__ b2,
                                             const float* __restrict__ g,
                                             const float* __restrict__ bln) {
  __shared__ __align__(16) float w1s[64][260];   // padded: async rows stay 16B aligned
  __shared__ __align__(16) float w2s[256][68];
  __shared__ __align__(16) float hid[16][260];
  __shared__ float yb[16][64];

  const int tid  = threadIdx.x;
  const int lane = tid & 31;
  const int wv   = tid >> 5;
  const int r    = lane & 15;
  const int half = lane >> 4;
  const int koff = half * 2;
  const int rowTile = blockIdx.x;

  // ---- stage w1 (64x256) and w2 (256x64) into LDS with async b128 copies
#pragma unroll
  for (int rnd = 0; rnd < 16; ++rnd) {
    int e4 = (rnd * 256 + tid) * 4;              // 4 floats per op, row-aligned
    async_copy_b128(lds_addr_of(&w1s[e4 >> 8][e4 & 255]), w1 + e4);
  }
#pragma unroll
  for (int rnd = 0; rnd < 16; ++rnd) {
    int e4 = (rnd * 256 + tid) * 4;
    async_copy_b128(lds_addr_of(&w2s[e4 >> 6][e4 & 63]), w2 + e4);
  }
  wait_async0();
  __syncthreads();

  // ---- phase 1: hidden = relu(h1 @ w1 + b1), 8 waves x 2 col-tiles
  {
    const float* arow = h1out + (size_t)(rowTile * 16 + r) * EMBED + koff;
    const int n0 = wv * 32 + r;
    v8f a0 = {}, a1 = {};
#pragma unroll
    for (int k0 = 0; k0 < EMBED; k0 += 4) {
      v2f a;  a.x  = arow[k0];                a.y  = arow[k0 + 1];
      v2f bA; bA.x = w1s[k0 + koff][n0];      bA.y = w1s[k0 + koff + 1][n0];
      v2f bB; bB.x = w1s[k0 + koff][n0 + 16]; bB.y = w1s[k0 + koff + 1][n0 + 16];
      a0 = wmma4(a, bA, a0);
      a1 = wmma4(a, bB, a1);
    }
    const int rl = half * 8;
    const float bb0 = b1[n0], bb1 = b1[n0 + 16];
#pragma unroll
    for (int i = 0; i < 8; ++i) {
      hid[rl + i][n0]      = fmaxf(a0[i] + bb0, 0.0f);
      hid[rl + i][n0 + 16] = fmaxf(a1[i] + bb1, 0.0f);
    }
  }
  __syncthreads();

  // ---- phase 2: f = hidden @ w2 + b2 ; y = h1 + f  (waves 0-3)
  if (wv < 4) {
    const int n = wv * 16 + r;
    v8f acc = {};
#pragma unroll 8
    for (int k0 = 0; k0 < FFND; k0 += 4) {
      v2f a; a.x = hid[r][k0 + koff];
             a.y = hid[r][k0 + koff + 1];
      v2f b; b.x = w2s[k0 + koff][n];
             b.y = w2s[k0 + koff + 1][n];
      acc = wmma4(a, b, acc);
    }
    const int rl = half * 8;
    const float bb = b2[n];
#pragma unroll
    for (int i = 0; i < 8; ++i) {
      int grow = rowTile * 16 + rl + i;
      yb[rl + i][n] = acc[i] + bb + h1out[(size_t)grow * EMBED + n];
    }
  }
  __syncthreads();

  // ---- LN2, one thread per row
  if (tid < 16) {
    int rr = tid;
    float s = 0.0f;
#pragma unroll
    for (int c = 0; c < 64; ++c) s += yb[rr][c];
    float mean = s * (1.0f / 64.0f);
    float v = 0.0f;
#pragma unroll
    for (int c = 0; c < 64; ++c) { float d = yb[rr][c] - mean; v += d * d; }
    float rs = rsqrtf(v * (1.0f / 64.0f) + LNEPS);
    float* o = h1out + (size_t)(rowTile * 16 + rr) * EMBED;
#pragma unroll
    for (int c = 0; c < 64; ++c) o[c] = (yb[rr][c] - mean) * rs * g[c] + bln[c];
  }
}

extern "C" void kernel_launch(void* const* d_in, const int* in_sizes, int n_in,
                              void* d_out, int out_size, void* d_ws, size_t ws_size,
                              hipStream_t stream) {
  const float* x      = (const float*)d_in[0];
  const int*   eidx   = (const int*)d_in[1];
  const float* attn_w = (const float*)d_in[2];
  const float* attn_b = (const float*)d_in[3];
  const float* w1     = (const float*)d_in[4];
  const float* b1     = (const float*)d_in[5];
  const float* w2     = (const float*)d_in[6];
  const float* b2     = (const float*)d_in[7];
  const float* ln1g   = (const float*)d_in[8];
  const float* ln1b   = (const float*)d_in[9];
  const float* ln2g   = (const float*)d_in[10];
  const float* ln2b   = (const float*)d_in[11];
  float* out = (float*)d_out;
  float* ws  = (float*)d_ws;

  const int N = NND;
  const int E = in_sizes[1] / 2;

  float* qkv = ws;                          // N*192 f32
  float* acc = qkv + (size_t)N * QKVC;      // N*64  f32
  int* counts = (int*)(acc + (size_t)N * EMBED); // N
  int* cursor = counts + N;                 // N (contiguous with counts for zeroing)
  int* basea  = cursor + N;                 // N
  int* elist  = basea + N;                  // E

  // 1) zero CSR counters
  k_zero_int<<<(2 * N + 255) / 256, 256, 0, stream>>>(counts, 2 * N);
  // 2) QKV projection (WMMA)
  k_qkv<<<(N / 16) * 12 / 8, 256, 0, stream>>>(x, attn_w, attn_b, qkv);
  // 3) CSR: histogram -> scan -> scatter
  k_count<<<(E + 255) / 256, 256, 0, stream>>>(eidx, E, counts);
  k_scan<<<1, 1024, 0, stream>>>(counts, basea, N);
  k_scatter<<<(E + 255) / 256, 256, 0, stream>>>(eidx, E, basea, cursor, elist);
  // 4) gather attention (no float atomics), normalized output in acc
  k_attn<<<(N * 8 + 255) / 256, 256, 0, stream>>>(eidx, E, counts, basea, elist,
                                                  qkv, acc);
  // 5) h1 = LN1(x + attn) -> d_out (reused as h1 scratch)
  k_ln1<<<N / 8, 256, 0, stream>>>(x, acc, ln1g, ln1b, out);
  // 6) fused FFN + LN2 (WMMA + async LDS weight staging), in place over d_out
  k_ffn<<<N / 16, 256, 0, stream>>>(out, w1, b1, w2, b2, ln2g, ln2b);
}